// GAT_81063212744722
// MI455X (gfx1250) — compile-verified
//
#include <hip/hip_runtime.h>
#include <hip/hip_bf16.h>

// GAT forward for MI455X (gfx1250, wave32).
// Bottleneck: ~1.4 GB edge gather/scatter traffic (~60us @ 23.3 TB/s).
// GEMMs (~2.5 GFLOP) on V_WMMA_F32_16X16X4_F32 with LDS-staged weights
// (ds_load_b64 B operands in exact WMMA lane layout -> no per-WMMA global
// load waits, no spills).

typedef __attribute__((ext_vector_type(2))) float v2f;
typedef __attribute__((ext_vector_type(8))) float v8f;

#define NEG_SLOPE 0.2f

// ---------------------------------------------------------------------------
// Zero init: workspace accumulation block + d_out
// ---------------------------------------------------------------------------
__global__ void gat_zero_kernel(float* __restrict__ a, size_t na,
                                float* __restrict__ b, int nb) {
    size_t i = (size_t)blockIdx.x * blockDim.x + threadIdx.x;
    size_t stride = (size_t)gridDim.x * blockDim.x;
    for (size_t k = i; k < na; k += stride) a[k] = 0.0f;
    if (i < (size_t)nb) b[i] = 0.0f;
}

// ---------------------------------------------------------------------------
// GEMM: C[M][ldn] = A[M][128] * B[128][ldn], fp32 WMMA 16x16x4.
// Block = 256 threads = 8 waves; W staged in LDS as K-pairs (float2) so each
// lane's B operand (VGPR v holds K = kk + v + 2*(lane/16)) is one ds_load_b64.
// Each wave owns one 16-row M-tile; NT = ldn/16 column tiles accumulate in
// registers (8*NT VGPRs).
// A lane layout (16x4 f32): row = lane%16, K pair at kk + 2*(lane/16) -> b64.
// C/D: VGPR r holds row r + 8*(lane/16), col = lane%16.
// ---------------------------------------------------------------------------
template <int NT>
__global__ __launch_bounds__(256)
void gat_gemm_wmma_f32(const float* __restrict__ A,
                       const float* __restrict__ B,
                       float* __restrict__ C, int M) {
    constexpr int K   = 128;
    constexpr int ldn = NT * 16;
    __shared__ float2 ldsW[(K / 2) * ldn];          // 32KB (NT=4) / 64KB (NT=8)

    const int tid = threadIdx.x;

    // Stage W as (K/2) x ldn pairs: ldsW[k2*ldn+col] = {W[2k2][col], W[2k2+1][col]}
    for (int idx = tid; idx < (K / 2) * ldn; idx += 256) {
        const int k2  = idx / ldn;
        const int col = idx - k2 * ldn;
        float2 p;
        p.x = B[(size_t)(2 * k2 + 0) * ldn + col];
        p.y = B[(size_t)(2 * k2 + 1) * ldn + col];
        ldsW[idx] = p;
    }
    __syncthreads();

    const int lane = tid & 31;
    const int wave = tid >> 5;
    const int half = lane >> 4;                     // 0 or 1
    const int l16  = lane & 15;
    const int mt   = blockIdx.x * 8 + wave;
    const int mtiles = (M + 15) >> 4;
    if (mt >= mtiles) return;                       // wave-uniform

    int arow = mt * 16 + l16;
    if (arow >= M) arow = M - 1;
    const float* __restrict__ Arow = A + (size_t)arow * K;

    v8f acc[NT];
#pragma unroll
    for (int t = 0; t < NT; ++t) acc[t] = (v8f){0.f,0.f,0.f,0.f,0.f,0.f,0.f,0.f};

#pragma unroll 4
    for (int kk = 0; kk < K; kk += 4) {
        v2f a = *(const v2f*)(Arow + kk + 2 * half);      // K = ka, ka+1
        if (kk + 8 < K) __builtin_prefetch(Arow + kk + 8, 0, 0);
        const float2* __restrict__ bp = &ldsW[(kk / 2 + half) * ldn + l16];
#pragma unroll
        for (int t = 0; t < NT; ++t) {
            float2 bv = bp[t * 16];                        // ds_load_b64
            v2f b;
            b[0] = bv.x;
            b[1] = bv.y;
            acc[t] = __builtin_amdgcn_wmma_f32_16x16x4_f32(
                false, a, false, b, (short)0, acc[t], false, false);
        }
    }

#pragma unroll
    for (int t = 0; t < NT; ++t) {
        const int col = t * 16 + l16;
#pragma unroll
        for (int r = 0; r < 8; ++r) {
            const int orow = mt * 16 + r + 8 * half;
            if (orow < M) C[(size_t)orow * ldn + col] = acc[t][r];
        }
    }
}

// ---------------------------------------------------------------------------
// Per-node attention scores: a_s[n,h] = <h[n,h,:], att_s[h,:]>, same for a_d.
// ---------------------------------------------------------------------------
__global__ void gat_att_scores(const float* __restrict__ h,
                               const float* __restrict__ att_s,
                               const float* __restrict__ att_d,
                               float* __restrict__ a_s,
                               float* __restrict__ a_d,
                               int n, int H, int F) {
    int i = blockIdx.x * blockDim.x + threadIdx.x;
    if (i >= n * H) return;
    int node = i / H, hh = i - node * H;
    const float* hp = h + (size_t)node * H * F + (size_t)hh * F;
    const float* sp = att_s + (size_t)hh * F;
    const float* dp = att_d + (size_t)hh * F;
    float s = 0.f, d = 0.f;
    for (int f = 0; f < F; ++f) {
        float v = hp[f];
        s += v * sp[f];
        d += v * dp[f];
    }
    a_s[i] = s;
    a_d[i] = d;
}

// ---------------------------------------------------------------------------
// Softmax denominator per (dst, head). One thread per edge (incl. self loops).
// Max-subtraction skipped: logits are small, alpha = e/denom is identical.
// ---------------------------------------------------------------------------
__global__ void gat_edge_denom(const int* __restrict__ ei, int E, int n, int H,
                               const float* __restrict__ a_s,
                               const float* __restrict__ a_d,
                               float* __restrict__ ssum) {
    int e = blockIdx.x * blockDim.x + threadIdx.x;
    int tot = E + n;
    if (e >= tot) return;
    int src = (e < E) ? ei[e]     : (e - E);
    int dst = (e < E) ? ei[E + e] : (e - E);
    for (int hh = 0; hh < H; ++hh) {
        float l = a_s[(size_t)src * H + hh] + a_d[(size_t)dst * H + hh];
        l = (l > 0.f) ? l : NEG_SLOPE * l;
        atomicAdd(&ssum[(size_t)dst * H + hh], __expf(l));
    }
}

// ---------------------------------------------------------------------------
// Weighted aggregation: one wave32 per edge; lanes stripe the H*F features.
// agg[dst] += h[src] * alpha  (global fp32 atomics, coalesced per 32 lanes).
// ---------------------------------------------------------------------------
template <int H, int F>
__global__ void gat_edge_aggregate(const int* __restrict__ ei, int E, int n,
                                   const float* __restrict__ a_s,
                                   const float* __restrict__ a_d,
                                   const float* __restrict__ ssum,
                                   const float* __restrict__ h,
                                   float* __restrict__ agg) {
    const int lane = threadIdx.x & 31;
    const int e    = blockIdx.x * (blockDim.x >> 5) + (threadIdx.x >> 5);
    const int tot  = E + n;
    if (e >= tot) return;
    const int src = (e < E) ? ei[e]     : (e - E);
    const int dst = (e < E) ? ei[E + e] : (e - E);

    float alpha[H];
#pragma unroll
    for (int hh = 0; hh < H; ++hh) {
        float l = a_s[(size_t)src * H + hh] + a_d[(size_t)dst * H + hh];
        l = (l > 0.f) ? l : NEG_SLOPE * l;
        alpha[hh] = __expf(l) / ssum[(size_t)dst * H + hh];
    }

    const float* hs = h   + (size_t)src * (H * F);
    float*       ag = agg + (size_t)dst * (H * F);
#pragma unroll
    for (int it = 0; it < (H * F) / 32; ++it) {
        const int j  = it * 32 + lane;
        const int hh = j / F;
        atomicAdd(&ag[j], hs[j] * alpha[hh]);
    }
}

// ---------------------------------------------------------------------------
// out = elu(agg + bias), in place.
// ---------------------------------------------------------------------------
__global__ void gat_bias_elu(float* __restrict__ a, const float* __restrict__ bias,
                             int n, int C) {
    int i = blockIdx.x * blockDim.x + threadIdx.x;
    if (i >= n * C) return;
    float v = a[i] + bias[i % C];
    a[i] = (v > 0.f) ? v : (__expf(v) - 1.f);
}

// ---------------------------------------------------------------------------
// pooled[c] = sum_n elu(agg2[n,c] + b2[c]); LDS partials then 64 atomics/block.
// ---------------------------------------------------------------------------
__global__ void gat_pool(const float* __restrict__ agg2,
                         const float* __restrict__ b2,
                         float* __restrict__ pooled, int n) {
    __shared__ float col[64];
    const int tid = threadIdx.x;
    if (tid < 64) col[tid] = 0.f;
    __syncthreads();

    const int c = tid & 63;
    const int rowsPerBlk = blockDim.x >> 6;   // 4 with blockDim 256
    const float bv = b2[c];
    float accv = 0.f;
    for (int r = blockIdx.x * rowsPerBlk + (tid >> 6); r < n;
         r += gridDim.x * rowsPerBlk) {
        float v = agg2[(size_t)r * 64 + c] + bv;
        accv += (v > 0.f) ? v : (__expf(v) - 1.f);
    }
    atomicAdd(&col[c], accv);
    __syncthreads();
    if (tid < 64) atomicAdd(&pooled[tid], col[tid]);
}

// ---------------------------------------------------------------------------
// out[j] = pooled @ Wl + bl   (64 -> 2)
// ---------------------------------------------------------------------------
__global__ void gat_head(const float* __restrict__ pooled,
                         const float* __restrict__ Wl,
                         const float* __restrict__ bl,
                         float* __restrict__ out2) {
    int j = threadIdx.x;
    if (j < 2) {
        float s = bl[j];
        for (int k = 0; k < 64; ++k) s += pooled[k] * Wl[k * 2 + j];
        out2[j] = s;
    }
}

// ---------------------------------------------------------------------------
extern "C" void kernel_launch(void* const* d_in, const int* in_sizes, int n_in,
                              void* d_out, int out_size, void* d_ws, size_t ws_size,
                              hipStream_t stream) {
    const float* x      = (const float*)d_in[0];
    const int*   ei     = (const int*)  d_in[1];
    const float* W1     = (const float*)d_in[2];
    const float* att_s1 = (const float*)d_in[3];
    const float* att_d1 = (const float*)d_in[4];
    const float* b1     = (const float*)d_in[5];
    const float* W2     = (const float*)d_in[6];
    const float* att_s2 = (const float*)d_in[7];
    const float* att_d2 = (const float*)d_in[8];
    const float* b2     = (const float*)d_in[9];
    const float* Wl     = (const float*)d_in[10];
    const float* bl     = (const float*)d_in[11];
    float* out = (float*)d_out;

    const int N = in_sizes[0] / 128;   // 50000
    const int E = in_sizes[1] / 2;     // 800000
    const int tot = E + N;             // with self loops

    // Workspace layout (floats): zero-block first, then scratch.
    float* ws   = (float*)d_ws;
    float* agg1 = ws;                          // N*128 (zeroed)
    float* agg2 = agg1 + (size_t)N * 128;      // N*64  (zeroed)
    float* s1   = agg2 + (size_t)N * 64;       // N*4   (zeroed)
    float* s2   = s1   + (size_t)N * 4;        // N     (zeroed)
    size_t zcount = (size_t)N * 197;
    float* h1   = s2   + (size_t)N;            // N*128
    float* h2   = h1   + (size_t)N * 128;      // N*64
    float* as1  = h2   + (size_t)N * 64;       // N*4
    float* ad1  = as1  + (size_t)N * 4;        // N*4
    float* as2  = ad1  + (size_t)N * 4;        // N
    float* ad2  = as2  + (size_t)N;            // N
    (void)ad2; (void)ws_size; (void)n_in; (void)out_size;

    // 0) zero accumulators + d_out (66 floats)
    gat_zero_kernel<<<2048, 256, 0, stream>>>(agg1, zcount, out, 66);

    const int mtiles = (N + 15) / 16;
    const int gemmBlocks = (mtiles + 7) / 8;

    // ---- Layer 1: H=4, F=32 ----
    // h1 = x @ W1   (no bias in linear; GAT bias applies post-aggregation)
    gat_gemm_wmma_f32<8><<<gemmBlocks, 256, 0, stream>>>(x, W1, h1, N);
    gat_att_scores<<<(N * 4 + 255) / 256, 256, 0, stream>>>(
        h1, att_s1, att_d1, as1, ad1, N, 4, 32);
    gat_edge_denom<<<(tot + 255) / 256, 256, 0, stream>>>(
        ei, E, N, 4, as1, ad1, s1);
    gat_edge_aggregate<4, 32><<<(tot + 7) / 8, 256, 0, stream>>>(
        ei, E, N, as1, ad1, s1, h1, agg1);
    gat_bias_elu<<<(N * 128 + 255) / 256, 256, 0, stream>>>(agg1, b1, N, 128);

    // ---- Layer 2: H=1, F=64 ----
    gat_gemm_wmma_f32<4><<<gemmBlocks, 256, 0, stream>>>(agg1, W2, h2, N);
    gat_att_scores<<<(N + 255) / 256, 256, 0, stream>>>(
        h2, att_s2, att_d2, as2, ad2, N, 1, 64);
    gat_edge_denom<<<(tot + 255) / 256, 256, 0, stream>>>(
        ei, E, N, 1, as2, ad2, s2);
    gat_edge_aggregate<1, 64><<<(tot + 7) / 8, 256, 0, stream>>>(
        ei, E, N, as2, ad2, s2, h2, agg2);

    // ---- Pool + head ----
    gat_pool<<<512, 256, 0, stream>>>(agg2, b2, out, N);       // pooled -> out[0..63]
    gat_head<<<1, 64, 0, stream>>>(out, Wl, bl, out + 64);     // out[64..65]
}